// SegImageGenerate_79070347919385
// MI455X (gfx1250) — compile-verified
//
#include <hip/hip_runtime.h>
#include <cstdint>
#include <cmath>

// ---------------------------------------------------------------------------
// Stripe-mask blend + 7x7 depthwise box blur (edge padding) + soft-edge select
// B=8, C=3, H=W=1024 f32.  Purely bandwidth-bound (~302 MB min traffic,
// ~13 us floor at 23.3 TB/s).  Uses CDNA5 async global->LDS loads to stage
// haloed tiles; masks are computed analytically (zero bytes).
// ---------------------------------------------------------------------------

#define H_IMG   1024
#define W_IMG   1024
#define TILE_H  32
#define TILE_W  128
#define HALO    3
#define IN_H    (TILE_H + 2 * HALO)   // 38
#define IN_W    (TILE_W + 2 * HALO)   // 134
#define LDSW    136                   // padded LDS row stride (floats)
#define NTHREADS 256                  // 8 wave32s

// CDNA5 async global -> LDS copy (4B per lane). VDST VGPR carries the LDS
// byte offset; generic shared pointers hold that offset in their low 32 bits.
__device__ __forceinline__ void async_g2l_b32(const float* lds_ptr,
                                              const float* gptr) {
  uint32_t lds_off = (uint32_t)(uintptr_t)lds_ptr;
  asm volatile("global_load_async_to_lds_b32 %0, %1, off"
               :: "v"(lds_off), "v"(gptr)
               : "memory");
}

__device__ __forceinline__ void wait_async_zero() {
  asm volatile("s_wait_asynccnt 0" ::: "memory");
}

__global__ __launch_bounds__(NTHREADS)
void seg_blur_kernel(const float* __restrict__ amb,
                     const float* __restrict__ fullimg,
                     const int*   __restrict__ p_slope,
                     const int*   __restrict__ p_num_lines,
                     const int*   __restrict__ p_start,
                     const int*   __restrict__ p_vh,
                     float*       __restrict__ out)
{
  __shared__ float sA[IN_H * LDSW];   // amb halo  -> seg (in place)
  __shared__ float sB[IN_H * LDSW];   // full halo -> horizontal partial sums

  const int tid   = threadIdx.x;
  const int row0  = blockIdx.y * TILE_H;
  const int col0  = blockIdx.x * TILE_W;
  const size_t pbase = (size_t)blockIdx.z * (size_t)(H_IMG * W_IMG);

  // Scalar params (uniform -> s_load through constant cache)
  const float s      = (float)(*p_slope);
  const int   nl     = *p_num_lines;
  const float start  = (float)(*p_start);
  const float vh     = (float)(*p_vh);
  const float swf    = (float)(H_IMG / nl);      // strip width
  const float inv_sw = 1.0f / swf;
  const float rcp_s  = 1.0f / s;
  const float thr    = vh * sqrtf(1.0f + s * s);

  // ---- Phase 0: async halo loads (edge-clamped addresses == replicate pad)
  for (int idx = tid; idx < IN_H * IN_W; idx += NTHREADS) {
    int r  = idx / IN_W;
    int c  = idx - r * IN_W;
    int gi = min(max(row0 + r - HALO, 0), H_IMG - 1);
    int gj = min(max(col0 + c - HALO, 0), W_IMG - 1);
    size_t g = pbase + (size_t)gi * W_IMG + (size_t)gj;
    async_g2l_b32(&sA[r * LDSW + c], amb + g);
    async_g2l_b32(&sB[r * LDSW + c], fullimg + g);
  }
  wait_async_zero();   // this wave's async DMAs have landed in LDS
  __syncthreads();     // all waves' DMAs visible workgroup-wide

  // ---- Phase 1: stripe blend (seg) in place in sA
  for (int idx = tid; idx < IN_H * IN_W; idx += NTHREADS) {
    int r  = idx / IN_W;
    int c  = idx - r * IN_W;
    float fi = (float)min(max(row0 + r - HALO, 0), H_IMG - 1);
    float fj = (float)min(max(col0 + c - HALO, 0), W_IMG - 1);
    float n  = floorf((fj + fi * rcp_s - start) * inv_sw);  // n >= 0
    bool  mask = (((int)n) & 1) == 0;
    float a = sA[r * LDSW + c];
    float f = sB[r * LDSW + c];
    sA[r * LDSW + c] = mask ? a : f;
  }
  __syncthreads();

  // ---- Phase 2: horizontal 7-tap into sB (full-img values are dead now)
  float* mid = sB;                       // stride TILE_W
  for (int idx = tid; idx < IN_H * TILE_W; idx += NTHREADS) {
    int r = idx >> 7;                    // TILE_W == 128
    int c = idx & (TILE_W - 1);
    const float* p = &sA[r * LDSW + c];
    mid[r * TILE_W + c] = p[0] + p[1] + p[2] + p[3] + p[4] + p[5] + p[6];
  }
  __syncthreads();

  // ---- Phase 3: vertical 7-tap, analytic mask2, select, coalesced store
  for (int idx = tid; idx < TILE_H * TILE_W; idx += NTHREADS) {
    int r = idx >> 7;
    int c = idx & (TILE_W - 1);
    const float* p = &mid[r * TILE_W + c];
    float blur = (p[0] + p[1 * TILE_W] + p[2 * TILE_W] + p[3 * TILE_W] +
                  p[4 * TILE_W] + p[5 * TILE_W] + p[6 * TILE_W]) * (1.0f / 49.0f);

    float fi = (float)(row0 + r);
    float fj = (float)(col0 + c);
    float n  = floorf((fj + fi * rcp_s - start) * inv_sw);
    float line = fi + s * fj;
    float d0 = fabsf(line - s * (start + n * swf));
    float d1 = fabsf(line - s * (start + (n + 1.0f) * swf));
    bool  m2 = (d0 <= thr) || (d1 <= thr);

    float seg = sA[(r + HALO) * LDSW + (c + HALO)];
    out[pbase + (size_t)(row0 + r) * W_IMG + (size_t)(col0 + c)] =
        m2 ? blur : seg;
  }
}

extern "C" void kernel_launch(void* const* d_in, const int* in_sizes, int n_in,
                              void* d_out, int out_size, void* d_ws, size_t ws_size,
                              hipStream_t stream) {
  const float* amb     = (const float*)d_in[0];
  const float* fullimg = (const float*)d_in[1];
  const int*   slope   = (const int*)d_in[2];
  const int*   nlines  = (const int*)d_in[3];
  const int*   start   = (const int*)d_in[4];
  const int*   vh      = (const int*)d_in[5];
  float*       out     = (float*)d_out;

  int planes = in_sizes[0] / (H_IMG * W_IMG);   // B*C = 24
  dim3 grid(W_IMG / TILE_W, H_IMG / TILE_H, planes);
  seg_blur_kernel<<<grid, NTHREADS, 0, stream>>>(
      amb, fullimg, slope, nlines, start, vh, out);
}